// SisoMmsePicDetector_64811056497378
// MI455X (gfx1250) — compile-verified
//
#include <hip/hip_runtime.h>
#include <math.h>

typedef float v2f __attribute__((ext_vector_type(2)));
typedef float v8f __attribute__((ext_vector_type(8)));

#define EPS_ 1e-4f
#define INV_SQRT10 0.31622776601683794f

// Element of the realified (2x2-block) H matrix: H_real[r][c], r in [0,16), c in [0,8)
// block for z = x+iy is [[x,-y],[y,x]]
__device__ __forceinline__ float hreal_elem(const float* __restrict__ hre,
                                            const float* __restrict__ him,
                                            int r, int c) {
  int m = r >> 1, i = r & 1, k = c >> 1, j = c & 1;
  float re = hre[m * 4 + k];
  float im = him[m * 4 + k];
  return j ? (i ? re : -im) : (i ? im : re);
}

__global__ __launch_bounds__(256) void mmse_pic_kernel(
    const float* __restrict__ yr, const float* __restrict__ yi,
    const float* __restrict__ hr, const float* __restrict__ hi,
    const float* __restrict__ llr_a, const float* __restrict__ no,
    float* __restrict__ out, int NI)
{
  // per-wave private LDS slices (no cross-wave sharing -> no __syncthreads)
  __shared__ __align__(16) float s_hre[8][4][32];     // [wave][inst][m*4+k]
  __shared__ __align__(16) float s_him[8][4][32];
  __shared__ float s_ev[8][4][4];                     // [wave][inst][k]
  __shared__ __align__(16) float s_G[8][4][16 * 17];  // realified Gram, padded pitch

  const int tid = threadIdx.x;
  const int w = tid >> 5;   // wave in block
  const int L = tid & 31;   // lane in wave
  const int g = L >> 3;     // instance slot within wave
  const int l = L & 7;      // row index m within instance

  const int instBase = (blockIdx.x * 8 + w) * 4;
  if (instBase + 4 > NI) return;   // wave-uniform guard
  const int inst = instBase + g;

  const float noise = no[0];

  // ---- loads: lane l owns row m=l of y and H ----
  const float y_r = yr[inst * 8 + l];
  const float y_i = yi[inst * 8 + l];
  const float4 h4r = ((const float4*)hr)[inst * 8 + l];
  const float4 h4i = ((const float4*)hi)[inst * 8 + l];
  float hre_[4] = {h4r.x, h4r.y, h4r.z, h4r.w};
  float him_[4] = {h4i.x, h4i.y, h4i.z, h4i.w};

  // ---- per-stream constellation statistics on lanes l<4 (k = l) ----
  float lv[4] = {0.f, 0.f, 0.f, 0.f};
  float PC[16], LP[16];
  float shr = 0.f, shi = 0.f, ev = 0.f;
  if (l < 4) {
    float4 l4 = ((const float4*)llr_a)[inst * 4 + l];
    lv[0] = l4.x; lv[1] = l4.y; lv[2] = l4.z; lv[3] = l4.w;
    float p0[4], q0[4], Kk = 0.f;
#pragma unroll
    for (int i = 0; i < 4; ++i) {
      p0[i] = 1.0f / (1.0f + __expf(lv[i]));   // 0.5*(1-tanh(l/2))
      q0[i] = 1.0f - p0[i];
      float al = fabsf(lv[i]);
      Kk += 0.5f * al + log1pf(__expf(-al));
    }
#pragma unroll
    for (int p = 0; p < 16; ++p) {
      const int a0 = (p >> 3) & 1, a1 = (p >> 2) & 1, a2 = (p >> 1) & 1, a3 = p & 1;
      float prob = (a0 ? q0[0] : p0[0]) * (a1 ? q0[1] : p0[1]) *
                   (a2 ? q0[2] : p0[2]) * (a3 ? q0[3] : p0[3]);
      PC[p] = prob;
      float dot = (a0 ? lv[0] : -lv[0]) + (a1 ? lv[1] : -lv[1]) +
                  (a2 ? lv[2] : -lv[2]) + (a3 ? lv[3] : -lv[3]);
      LP[p] = 0.5f * dot - Kk;
      float re_p = (1.0f - 2.0f * a0) * (2.0f - (1.0f - 2.0f * a2)) * INV_SQRT10;
      float im_p = (1.0f - 2.0f * a1) * (2.0f - (1.0f - 2.0f * a3)) * INV_SQRT10;
      shr += re_p * prob;
      shi += im_p * prob;
    }
#pragma unroll
    for (int p = 0; p < 16; ++p) {
      const int a0 = (p >> 3) & 1, a1 = (p >> 2) & 1, a2 = (p >> 1) & 1, a3 = p & 1;
      float re_p = (1.0f - 2.0f * a0) * (2.0f - (1.0f - 2.0f * a2)) * INV_SQRT10;
      float im_p = (1.0f - 2.0f * a1) * (2.0f - (1.0f - 2.0f * a3)) * INV_SQRT10;
      float dr = shr - re_p, di = shi - im_p;
      ev += fmaxf(dr * dr + di * di, EPS_ * EPS_) * PC[p];
    }
    s_ev[w][g][l] = ev;
  }

  // stage H rows into LDS for the wave-wide WMMA operand build
  *(float4*)&s_hre[w][g][l * 4] = h4r;
  *(float4*)&s_him[w][g][l * 4] = h4i;
  asm volatile("s_wait_dscnt 0x0" ::: "memory");

  // ---- Gram matrices via WMMA: G_real(16x16) = H_real(16x8) * diag(ev2) * H_real^T ----
  {
    const int half = L >> 4;
    const int rc = L & 15;  // rowA for operand A, column n for operand B / C
#pragma unroll
    for (int gi = 0; gi < 4; ++gi) {
      const float* hreP = s_hre[w][gi];
      const float* himP = s_him[w][gi];
      const float* evP = s_ev[w][gi];
      v8f c = {0.f, 0.f, 0.f, 0.f, 0.f, 0.f, 0.f, 0.f};
#pragma unroll
      for (int s = 0; s < 2; ++s) {
        const int kb = 4 * s + 2 * half;  // ISA 16x4 f32 A layout: v0 K={kb}, v1 K={kb+1}
        v2f a, b;
        a.x = hreal_elem(hreP, himP, rc, kb);
        a.y = hreal_elem(hreP, himP, rc, kb + 1);
        b.x = evP[kb >> 1] * hreal_elem(hreP, himP, rc, kb);
        b.y = evP[(kb + 1) >> 1] * hreal_elem(hreP, himP, rc, kb + 1);
        c = __builtin_amdgcn_wmma_f32_16x16x4_f32(false, a, false, b,
                                                  (short)0, c, false, false);
      }
      float* Gg = s_G[w][gi];
#pragma unroll
      for (int j = 0; j < 8; ++j) Gg[(j + 8 * half) * 17 + rc] = c[j];
    }
  }
  asm volatile("s_wait_dscnt 0x0" ::: "memory");

  // ---- gather complex row m=l of A (+ noise*I) ----
  float ar[8], ai[8];
  {
    const float* Gm = s_G[w][g];
#pragma unroll
    for (int n = 0; n < 8; ++n) {
      ar[n] = Gm[(2 * l) * 17 + 2 * n] + ((n == l) ? noise : 0.0f);
      ai[n] = Gm[(2 * l + 1) * 17 + 2 * n];
    }
  }
  float rr[4], ri[4];
#pragma unroll
  for (int k = 0; k < 4; ++k) { rr[k] = hre_[k]; ri[k] = him_[k]; }

  // ---- row-owned pivotless complex Gaussian elimination (A Hermitian PD) ----
#pragma unroll
  for (int j = 0; j < 8; ++j) {
    float pr[8], pm[8], qr[4], qi[4];
#pragma unroll
    for (int n = j; n < 8; ++n) {
      pr[n] = __shfl(ar[n], j, 8);
      pm[n] = __shfl(ai[n], j, 8);
    }
#pragma unroll
    for (int k = 0; k < 4; ++k) {
      qr[k] = __shfl(rr[k], j, 8);
      qi[k] = __shfl(ri[k], j, 8);
    }
    const float inv = 1.0f / (pr[j] * pr[j] + pm[j] * pm[j] + 1e-30f);
    if (l > j) {
      const float fr = (ar[j] * pr[j] + ai[j] * pm[j]) * inv;
      const float fi = (ai[j] * pr[j] - ar[j] * pm[j]) * inv;
#pragma unroll
      for (int n = j; n < 8; ++n) {
        ar[n] -= fr * pr[n] - fi * pm[n];
        ai[n] -= fr * pm[n] + fi * pr[n];
      }
#pragma unroll
      for (int k = 0; k < 4; ++k) {
        rr[k] -= fr * qr[k] - fi * qi[k];
        ri[k] -= fr * qi[k] + fi * qr[k];
      }
    }
  }
  // back-substitution
#pragma unroll
  for (int n = 7; n >= 0; --n) {
    const float dr = __shfl(ar[n], n, 8);
    const float di = __shfl(ai[n], n, 8);
    const float inv = 1.0f / (dr * dr + di * di + 1e-30f);
#pragma unroll
    for (int k = 0; k < 4; ++k) {
      const float br = __shfl(rr[k], n, 8);
      const float bi = __shfl(ri[k], n, 8);
      const float xr = (br * dr + bi * di) * inv;
      const float xi = (bi * dr - br * di) * inv;
      if (l == n)      { rr[k] = xr; ri[k] = xi; }
      else if (l < n)  { rr[k] -= ar[n] * xr - ai[n] * xi;
                         ri[k] -= ar[n] * xi + ai[n] * xr; }
    }
  }
  // rr/ri now hold Ah[m][k] for row m=l

  // ---- mu[k] = Re sum_m conj(h) * Ah ----
  float mu[4];
#pragma unroll
  for (int k = 0; k < 4; ++k) mu[k] = hre_[k] * rr[k] + him_[k] * ri[k];
#pragma unroll
  for (int off = 1; off < 8; off <<= 1)
#pragma unroll
    for (int k = 0; k < 4; ++k) mu[k] += __shfl_xor(mu[k], off, 8);

  // ---- broadcast s_hat, residual, wHy ----
  float sr[4], si_[4];
#pragma unroll
  for (int k = 0; k < 4; ++k) { sr[k] = __shfl(shr, k, 8); si_[k] = __shfl(shi, k, 8); }
  float yrr = y_r, yri = y_i;
#pragma unroll
  for (int k = 0; k < 4; ++k) {
    yrr -= hre_[k] * sr[k] - him_[k] * si_[k];
    yri -= hre_[k] * si_[k] + him_[k] * sr[k];
  }
  float wr[4], wi[4];
#pragma unroll
  for (int k = 0; k < 4; ++k) {
    wr[k] = rr[k] * yrr + ri[k] * yri;   // conj(Ah) * y_res
    wi[k] = rr[k] * yri - ri[k] * yrr;
  }
#pragma unroll
  for (int off = 1; off < 8; off <<= 1)
#pragma unroll
    for (int k = 0; k < 4; ++k) {
      wr[k] += __shfl_xor(wr[k], off, 8);
      wi[k] += __shfl_xor(wi[k], off, 8);
    }

  // ---- final LLRs on lanes l<4 ----
  if (l < 4) {
    const float muk = (l == 0) ? mu[0] : (l == 1) ? mu[1] : (l == 2) ? mu[2] : mu[3];
    const float wrk = (l == 0) ? wr[0] : (l == 1) ? wr[1] : (l == 2) ? wr[2] : wr[3];
    const float wik = (l == 0) ? wi[0] : (l == 1) ? wi[1] : (l == 2) ? wi[2] : wi[3];
    const float zr = wrk / muk + shr;
    const float zi = wik / muk + shi;
    const float rho = muk / fmaxf(1.0f - ev * muk, EPS_);

    float E[16];
    float m1[4] = {-3.4e38f, -3.4e38f, -3.4e38f, -3.4e38f};
    float m0[4] = {-3.4e38f, -3.4e38f, -3.4e38f, -3.4e38f};
#pragma unroll
    for (int p = 0; p < 16; ++p) {
      const int a0 = (p >> 3) & 1, a1 = (p >> 2) & 1, a2 = (p >> 1) & 1, a3 = p & 1;
      float re_p = (1.0f - 2.0f * a0) * (2.0f - (1.0f - 2.0f * a2)) * INV_SQRT10;
      float im_p = (1.0f - 2.0f * a1) * (2.0f - (1.0f - 2.0f * a3)) * INV_SQRT10;
      float dr = zr - re_p, di = zi - im_p;
      E[p] = -fmaxf(dr * dr + di * di, EPS_ * EPS_) * rho + LP[p];
#pragma unroll
      for (int i = 0; i < 4; ++i) {
        const int bit = (p >> (3 - i)) & 1;
        if (bit) m1[i] = fmaxf(m1[i], E[p]); else m0[i] = fmaxf(m0[i], E[p]);
      }
    }
    float s1[4] = {0.f, 0.f, 0.f, 0.f}, s0[4] = {0.f, 0.f, 0.f, 0.f};
#pragma unroll
    for (int p = 0; p < 16; ++p) {
#pragma unroll
      for (int i = 0; i < 4; ++i) {
        const int bit = (p >> (3 - i)) & 1;
        if (bit) s1[i] += __expf(E[p] - m1[i]); else s0[i] += __expf(E[p] - m0[i]);
      }
    }
    float4 o;
    float od[4];
#pragma unroll
    for (int i = 0; i < 4; ++i)
      od[i] = (m1[i] + __logf(s1[i])) - (m0[i] + __logf(s0[i])) - lv[i];
    o.x = od[0]; o.y = od[1]; o.z = od[2]; o.w = od[3];
    ((float4*)out)[inst * 4 + l] = o;
  }
}

extern "C" void kernel_launch(void* const* d_in, const int* in_sizes, int n_in,
                              void* d_out, int out_size, void* d_ws, size_t ws_size,
                              hipStream_t stream) {
  const float* yr = (const float*)d_in[0];
  const float* yi = (const float*)d_in[1];
  const float* hr = (const float*)d_in[2];
  const float* hi = (const float*)d_in[3];
  const float* llr = (const float*)d_in[4];
  const float* no = (const float*)d_in[5];
  float* out = (float*)d_out;

  const int NI = in_sizes[0] / 8;            // B*T*S instances (M=8)
  const int blocks = (NI + 31) / 32;         // 32 instances per 256-thread block
  mmse_pic_kernel<<<blocks, 256, 0, stream>>>(yr, yi, hr, hi, llr, no, out, NI);
}